// RoIAlign_43241730736872
// MI455X (gfx1250) — compile-verified
//
#include <hip/hip_runtime.h>

// RoIAlign (bilinear, 7x7 grid, (AH-1)-bin style), f32.
// features: (2, 256, 64, 64) NCHW f32   rois: (8000, 5) f32 [b, x1, y1, x2, y2]
// out: (8000, 256, 7, 7) f32
//
// Roofline: 401 MB of output stores -> ~17 us floor @ 23.3 TB/s HBM; the 8 MB
// feature map is L2-resident so gathers are L2 traffic, not HBM. Kernel is
// built to hit the store roofline: one block per ROI, coalesced NT b32 stores,
// 2x b64 gathers per element (paired bilinear corners), per-ROI coefficient
// table staged once in LDS, ROI record fetched via the CDNA5 async
// global->LDS path (global_load_async_to_lds_b32 / s_wait_asynccnt), and all
// per-element address math reduced to one 32-bit v_add against an SGPR base.

namespace {
constexpr int   kAH   = 7;
constexpr int   kAW   = 7;
constexpr int   kNP   = kAH * kAW;       // 49 sample points
constexpr int   kC    = 256;
constexpr int   kH    = 64;
constexpr int   kW    = 64;
constexpr int   kPlane = kH * kW;        // 4096 elements / channel plane
constexpr int   kImg   = kC * kPlane;    // 1048576 elements / batch image
constexpr int   kOutPerRoi = kC * kNP;   // 12544
constexpr float kScale = 0.0625f;
}

typedef float v2f  __attribute__((ext_vector_type(2)));
typedef v2f   v2fu __attribute__((aligned(4)));     // allow 4B-aligned b64 load
typedef float v4f  __attribute__((ext_vector_type(4)));

__global__ __launch_bounds__(256) void roialign_kernel(
    const float* __restrict__ feat,
    const float* __restrict__ rois,
    float* __restrict__ out)
{
    __shared__ float s_roi[5];
    __shared__ int   s_off[kNP];   // byte offset (h0*W + w0)*4 within a plane
    __shared__ v4f   s_wgt[kNP];   // bilinear weights (validity folded in)

    const int n   = blockIdx.x;
    const int tid = threadIdx.x;

    // ---- CDNA5 async global->LDS: stage this block's ROI record (5 x f32) ----
    if (tid < 5) {
        unsigned lds_b = (unsigned)(unsigned long long)(&s_roi[0]) + (unsigned)tid * 4u;
        const float* g = rois + (size_t)n * 5u + (size_t)tid;
        asm volatile("global_load_async_to_lds_b32 %0, %1, off"
                     :: "v"(lds_b), "v"(g) : "memory");
        asm volatile("s_wait_asynccnt 0" ::: "memory");
    }
    __syncthreads();

    // ---- Per-ROI bilinear table: 49 points -> (byte offset, 4 weights) ----
    if (tid < kNP) {
        const float x1 = s_roi[1] * kScale, y1 = s_roi[2] * kScale;
        const float x2 = s_roi[3] * kScale, y2 = s_roi[4] * kScale;
        const float bh = fmaxf(y2 - y1, 0.0f) / (float)(kAH - 1);
        const float bw = fmaxf(x2 - x1, 0.0f) / (float)(kAW - 1);
        const int   ph = tid / kAW;
        const int   pw = tid - ph * kAW;
        const float h  = y1 + (float)ph * bh;
        const float w  = x1 + (float)pw * bw;
        const bool valid = (h >= 0.0f) && (h < (float)kH) &&
                           (w >= 0.0f) && (w < (float)kW);
        const int h0 = (int)fminf(fmaxf(floorf(h), 0.0f), (float)(kH - 2));
        const int w0 = (int)fminf(fmaxf(floorf(w), 0.0f), (float)(kW - 2));
        const float lh = h - (float)h0;
        const float lw = w - (float)w0;
        const float vm = valid ? 1.0f : 0.0f;
        v4f wv;
        wv.x = (1.0f - lh) * (1.0f - lw) * vm;   // v00
        wv.y = (1.0f - lh) * lw          * vm;   // v01
        wv.z = lh          * (1.0f - lw) * vm;   // v10
        wv.w = lh          * lw          * vm;   // v11
        s_off[tid] = (h0 * kW + w0) * 4;         // pre-scaled to bytes
        s_wgt[tid] = wv;
    }
    __syncthreads();

    // Batch-image base in BYTES, forced uniform -> SGPR, so every gather is
    // saddr-form (SGPR base + one 32-bit voffset VGPR).
    const int baseB = __builtin_amdgcn_readfirstlane(((int)s_roi[0]) * (kImg * 4));
    const char* __restrict__ fb = (const char*)feat + baseB;
    float* __restrict__ ob = out + (size_t)n * kOutPerRoi + tid;

    // e = tid + i*256 ; c = e/49 ; p = e%49 (256 = 5*49 + 11).
    // coffB = c * plane bytes (c<<14), tracked incrementally.
    int p     = tid % kNP;
    int coffB = (tid / kNP) << 14;
    #pragma unroll 7
    for (int i = 0; i < kNP; ++i) {
        const int offB = s_off[p];
        const v4f wv   = s_wgt[p];
        const char* a  = fb + (coffB + offB);
        const v2f t = *(const v2fu*)(a);              // (v00, v01) adjacent
        const v2f b = *(const v2fu*)(a + kW * 4);     // (v10, v11) adjacent
        float r = t.x * wv.x;
        r = fmaf(t.y, wv.y, r);
        r = fmaf(b.x, wv.z, r);
        r = fmaf(b.y, wv.w, r);
        __builtin_nontemporal_store(r, ob + i * 256); // coalesced NT stream

        p += 11;
        const bool carry = (p >= kNP);
        coffB += carry ? (6 << 14) : (5 << 14);
        p     -= carry ? kNP : 0;
    }
}

extern "C" void kernel_launch(void* const* d_in, const int* in_sizes, int n_in,
                              void* d_out, int out_size, void* d_ws, size_t ws_size,
                              hipStream_t stream) {
    const float* features = (const float*)d_in[0];
    const float* rois     = (const float*)d_in[1];
    float* out            = (float*)d_out;
    const int N = in_sizes[1] / 5;   // 8000 ROIs
    roialign_kernel<<<dim3(N), dim3(256), 0, stream>>>(features, rois, out);
}